// E3NN_Transfer_Model_52510270161539
// MI455X (gfx1250) — compile-verified
//
#include <hip/hip_runtime.h>
#include <math.h>

#define D_FEAT   64
#define EDGE_DIM 16
#define HID      32
#define LOG2C    0.6931471805599453f

typedef __attribute__((ext_vector_type(2))) float v2f;
typedef __attribute__((ext_vector_type(8))) float v8f;

// Full-precision CDNA5 WMMA: D(16x16,f32) = A(16x4,f32) x B(4x16,f32) + C
__device__ __forceinline__ v8f wmma_k4(v2f a, v2f b, v8f c) {
  return __builtin_amdgcn_wmma_f32_16x16x4_f32(
      /*neg_a=*/false, a, /*neg_b=*/false, b,
      /*c_mod=*/(short)0, c, /*reuse_a=*/false, /*reuse_b=*/false);
}

// wave32 lockstep: a DS-counter wait is enough for intra-wave LDS RAW.
__device__ __forceinline__ void lds_fence() {
  asm volatile("s_wait_dscnt 0" ::: "memory");
}

__device__ __forceinline__ void atomic_add_f32(float* p, float v) {
  (void)__hip_atomic_fetch_add(p, v, __ATOMIC_RELAXED, __HIP_MEMORY_SCOPE_AGENT);
}

// silu with hardware v_rcp_f32 (avoid the correctly-rounded div sequence)
__device__ __forceinline__ float fast_silu(float x) {
  return x * __builtin_amdgcn_rcpf(1.0f + __expf(-x));
}

// shifted softplus with hardware v_log_f32 (avoid libm log1pf)
__device__ __forceinline__ float softplus_m(float x) {
  return fmaxf(x, 0.0f) + __logf(1.0f + __expf(-fabsf(x))) - LOG2C;
}

__device__ __forceinline__ float fast_elu(float x) {
  return x > 0.0f ? x : (__expf(x) - 1.0f);
}

// ---------------- utility kernels ----------------
__global__ void zero_kernel(float* __restrict__ p, int n) {
  int i = blockIdx.x * blockDim.x + threadIdx.x;
  if (i < n) p[i] = 0.0f;
}

__global__ void copy_kernel(const float* __restrict__ s, float* __restrict__ d, int n) {
  int i = blockIdx.x * blockDim.x + threadIdx.x;
  if (i < n) d[i] = s[i];
}

__global__ void count_deg_kernel(const int* __restrict__ dst, float* __restrict__ deg, int E) {
  int e = blockIdx.x * blockDim.x + threadIdx.x;
  if (e < E) atomic_add_f32(&deg[dst[e]], 1.0f);
}

__global__ void inv_deg_kernel(float* __restrict__ deg, int n) {
  int i = blockIdx.x * blockDim.x + threadIdx.x;
  if (i < n) { float d = deg[i]; deg[i] = (d > 0.5f) ? (1.0f / d) : 0.0f; }
}

// ---------------- fused edge kernel ----------------
// One wave == one tile of 16 edges.
//   GEMM1: edge_attr[16x16] @ rw1[16x32] -> silu -> H (LDS)
//   GEMM2: H[16x32] @ rw2[32x64] + rb2   -> R (LDS)
//   scatter: agg[dst] += h[src] * R   (coalesced float2 loads, hw f32 atomics)
__global__ __launch_bounds__(256) void edge_kernel(
    const float* __restrict__ h,  const float* __restrict__ ea,
    const int*  __restrict__ src, const int*  __restrict__ dst,
    const float* __restrict__ rw1, const float* __restrict__ rb1,
    const float* __restrict__ rw2, const float* __restrict__ rb2,
    float* __restrict__ agg, int ntiles)
{
  __shared__ float ldsH[8][16][34];   // padded: 34*r distinct banks
  __shared__ float ldsR[8][16][66];   // padded, even stride for v2f reads

  const int wave = threadIdx.x >> 5;
  const int lane = threadIdx.x & 31;
  const int tile = blockIdx.x * 8 + wave;
  if (tile >= ntiles) return;
  const int eb  = tile * 16;
  const int m   = lane & 15;                 // A-row this lane serves
  const int kof = (lane >= 16) ? 2 : 0;      // K sub-offset per lane half
  const int n4  = lane & 15;                 // B/C column within 16-wide tile
  const int mo  = (lane >= 16) ? 8 : 0;      // C-layout row offset

  // ---- GEMM1: K = 16 -> 4 wmma steps, two 16-col accumulators ----
  v8f c0 = {}; v8f c1 = {};
  const float* earow = ea + (size_t)(eb + m) * EDGE_DIM;
  #pragma unroll
  for (int kk = 0; kk < 4; ++kk) {
    const int k = kk * 4 + kof;
    v2f a; a.x = earow[k]; a.y = earow[k + 1];
    v2f b;
    b.x = rw1[k * HID + n4];        b.y = rw1[(k + 1) * HID + n4];
    c0 = wmma_k4(a, b, c0);
    b.x = rw1[k * HID + 16 + n4];   b.y = rw1[(k + 1) * HID + 16 + n4];
    c1 = wmma_k4(a, b, c1);
  }
  #pragma unroll
  for (int i = 0; i < 8; ++i) {
    const int M = i + mo;
    ldsH[wave][M][n4]      = fast_silu(c0[i] + rb1[n4]);
    ldsH[wave][M][16 + n4] = fast_silu(c1[i] + rb1[16 + n4]);
  }
  lds_fence();

  // ---- GEMM2: K = 32 -> 8 wmma steps, four 16-col accumulators ----
  v8f d0 = {}, d1 = {}, d2 = {}, d3 = {};
  #pragma unroll
  for (int kk = 0; kk < 8; ++kk) {
    const int k = kk * 4 + kof;
    v2f a; a.x = ldsH[wave][m][k]; a.y = ldsH[wave][m][k + 1];
    const float* B0 = rw2 + (size_t)k * D_FEAT;
    const float* B1 = rw2 + (size_t)(k + 1) * D_FEAT;
    v2f b;
    b.x = B0[n4];      b.y = B1[n4];      d0 = wmma_k4(a, b, d0);
    b.x = B0[16+n4];   b.y = B1[16+n4];   d1 = wmma_k4(a, b, d1);
    b.x = B0[32+n4];   b.y = B1[32+n4];   d2 = wmma_k4(a, b, d2);
    b.x = B0[48+n4];   b.y = B1[48+n4];   d3 = wmma_k4(a, b, d3);
  }
  #pragma unroll
  for (int i = 0; i < 8; ++i) {
    const int M = i + mo;
    ldsR[wave][M][n4]      = d0[i] + rb2[n4];
    ldsR[wave][M][16 + n4] = d1[i] + rb2[16 + n4];
    ldsR[wave][M][32 + n4] = d2[i] + rb2[32 + n4];
    ldsR[wave][M][48 + n4] = d3[i] + rb2[48 + n4];
  }
  lds_fence();

  // ---- gather + modulate + scatter-add ----
  for (int e = 0; e < 16; ++e) {
    const int s = src[eb + e];
    const int d = dst[eb + e];
    const float* hrow = h + (size_t)s * D_FEAT;
    float* arow = agg + (size_t)d * D_FEAT;
    const v2f hv = *(const v2f*)(hrow + 2 * lane);          // coalesced 256B/wave
    const v2f rv = *(const v2f*)(&ldsR[wave][e][2 * lane]);
    atomic_add_f32(&arow[2 * lane],     hv.x * rv.x);
    atomic_add_f32(&arow[2 * lane + 1], hv.y * rv.y);
  }
}

// scalar fallback for edge tail (E % 16)
__global__ void edge_tail_kernel(
    const float* __restrict__ h,  const float* __restrict__ ea,
    const int*  __restrict__ src, const int*  __restrict__ dst,
    const float* __restrict__ rw1, const float* __restrict__ rb1,
    const float* __restrict__ rw2, const float* __restrict__ rb2,
    float* __restrict__ agg, int e0, int E)
{
  int e = e0 + blockIdx.x * blockDim.x + threadIdx.x;
  if (e >= E) return;
  float hid[HID];
  const float* a = ea + (size_t)e * EDGE_DIM;
  for (int j = 0; j < HID; ++j) {
    float acc = rb1[j];
    for (int k = 0; k < EDGE_DIM; ++k) acc += a[k] * rw1[k * HID + j];
    hid[j] = fast_silu(acc);
  }
  const int s = src[e], d = dst[e];
  const float* hrow = h + (size_t)s * D_FEAT;
  float* arow = agg + (size_t)d * D_FEAT;
  for (int j = 0; j < D_FEAT; ++j) {
    float acc = rb2[j];
    for (int k = 0; k < HID; ++k) acc += hid[k] * rw2[k * D_FEAT + j];
    atomic_add_f32(&arow[j], hrow[j] * acc);
  }
}

// ---------------- node self-interaction GEMM ----------------
// h = softplus((agg*inv_deg) @ lw + lb) - ln2 ; one wave per 16-node tile.
__global__ __launch_bounds__(256) void node_kernel(
    const float* __restrict__ agg, const float* __restrict__ inv_deg,
    const float* __restrict__ lw,  const float* __restrict__ lb,
    float* __restrict__ h, int ntiles)
{
  const int wave = threadIdx.x >> 5;
  const int lane = threadIdx.x & 31;
  const int tile = blockIdx.x * 8 + wave;
  if (tile >= ntiles) return;
  const int nb  = tile * 16;
  const int m   = lane & 15;
  const int kof = (lane >= 16) ? 2 : 0;
  const int n4  = lane & 15;
  const int mo  = (lane >= 16) ? 8 : 0;

  const float w = inv_deg[nb + m];                 // fold scatter-mean into A
  const float* arow = agg + (size_t)(nb + m) * D_FEAT;

  v8f d0 = {}, d1 = {}, d2 = {}, d3 = {};
  #pragma unroll
  for (int kk = 0; kk < 16; ++kk) {
    const int k = kk * 4 + kof;
    v2f a; a.x = arow[k] * w; a.y = arow[k + 1] * w;
    const float* B0 = lw + (size_t)k * D_FEAT;
    const float* B1 = lw + (size_t)(k + 1) * D_FEAT;
    v2f b;
    b.x = B0[n4];      b.y = B1[n4];      d0 = wmma_k4(a, b, d0);
    b.x = B0[16+n4];   b.y = B1[16+n4];   d1 = wmma_k4(a, b, d1);
    b.x = B0[32+n4];   b.y = B1[32+n4];   d2 = wmma_k4(a, b, d2);
    b.x = B0[48+n4];   b.y = B1[48+n4];   d3 = wmma_k4(a, b, d3);
  }
  #pragma unroll
  for (int i = 0; i < 8; ++i) {
    float* hrow = h + (size_t)(nb + i + mo) * D_FEAT;
    hrow[n4]      = softplus_m(d0[i] + lb[n4]);
    hrow[16 + n4] = softplus_m(d1[i] + lb[16 + n4]);
    hrow[32 + n4] = softplus_m(d2[i] + lb[32 + n4]);
    hrow[48 + n4] = softplus_m(d3[i] + lb[48 + n4]);
  }
}

__global__ void node_tail_kernel(
    const float* __restrict__ agg, const float* __restrict__ inv_deg,
    const float* __restrict__ lw,  const float* __restrict__ lb,
    float* __restrict__ h, int n0, int N)
{
  int n = n0 + blockIdx.x * blockDim.x + threadIdx.x;
  if (n >= N) return;
  const float w = inv_deg[n];
  const float* arow = agg + (size_t)n * D_FEAT;
  float* hrow = h + (size_t)n * D_FEAT;
  for (int j = 0; j < D_FEAT; ++j) {
    float acc = lb[j];
    for (int k = 0; k < D_FEAT; ++k) acc += arow[k] * w * lw[k * D_FEAT + j];
    hrow[j] = softplus_m(acc);
  }
}

// ---------------- pooling + head ----------------
__global__ void pool_kernel(const float* __restrict__ h, const int* __restrict__ batch,
                            float* __restrict__ gsum, float* __restrict__ gcnt, int n_elem)
{
  int idx = blockIdx.x * blockDim.x + threadIdx.x;
  if (idx >= n_elem) return;
  const int node = idx >> 6, c = idx & 63;
  const int g = batch[node];
  atomic_add_f32(&gsum[g * D_FEAT + c], h[idx]);
  if (c == 0) atomic_add_f32(&gcnt[g], 1.0f);
}

__global__ __launch_bounds__(256) void head_kernel(
    const float* __restrict__ gsum, const float* __restrict__ gcnt,
    const float* __restrict__ fw0, const float* __restrict__ fb0,
    const float* __restrict__ fw1, const float* __restrict__ fb1,
    const float* __restrict__ fw2, const float* __restrict__ fb2,
    float* __restrict__ out, int G)
{
  __shared__ float P[64][64];     // pooled, later reused for z1
  __shared__ float Z0[64][128];
  const int tid = threadIdx.x;

  for (int idx = tid; idx < G * 64; idx += 256) {
    const int g = idx >> 6, c = idx & 63;
    P[g][c] = gsum[idx] / fmaxf(gcnt[g], 1.0f);
  }
  __syncthreads();
  for (int idx = tid; idx < G * 128; idx += 256) {
    const int g = idx >> 7, j = idx & 127;
    float acc = fb0[j];
    for (int k = 0; k < 64; ++k) acc += P[g][k] * fw0[k * 128 + j];
    Z0[g][j] = fast_elu(acc);
  }
  __syncthreads();
  for (int idx = tid; idx < G * 64; idx += 256) {
    const int g = idx >> 6, j = idx & 63;
    float acc = fb1[j];
    for (int k = 0; k < 128; ++k) acc += Z0[g][k] * fw1[k * 64 + j];
    P[g][j] = fast_elu(acc);                 // z1 overwrites P
  }
  __syncthreads();
  for (int g = tid; g < G; g += 256) {
    float acc = fb2[0];
    for (int k = 0; k < 64; ++k) acc += P[g][k] * fw2[k];
    out[g] = acc;
  }
}

// ---------------- launcher ----------------
extern "C" void kernel_launch(void* const* d_in, const int* in_sizes, int n_in,
                              void* d_out, int out_size, void* d_ws, size_t ws_size,
                              hipStream_t stream) {
  const float* x    = (const float*)d_in[0];
  const int*   ei   = (const int*)  d_in[1];
  const float* ea   = (const float*)d_in[2];
  const int*   batch= (const int*)  d_in[3];
  const float* rw1  = (const float*)d_in[4];
  const float* rb1  = (const float*)d_in[5];
  const float* rw2  = (const float*)d_in[6];
  const float* rb2  = (const float*)d_in[7];
  const float* lw   = (const float*)d_in[8];
  const float* lb   = (const float*)d_in[9];
  const float* fw0  = (const float*)d_in[10];
  const float* fb0  = (const float*)d_in[11];
  const float* fw1  = (const float*)d_in[12];
  const float* fb1  = (const float*)d_in[13];
  const float* fw2  = (const float*)d_in[14];
  const float* fb2  = (const float*)d_in[15];
  float* out = (float*)d_out;

  const int N = in_sizes[0] / D_FEAT;
  const int E = in_sizes[1] / 2;
  const int G = out_size;
  const int* src = ei;
  const int* dst = ei + E;

  // workspace layout
  float* ws   = (float*)d_ws;
  float* h    = ws;                              // N*64
  float* agg  = h   + (size_t)N * D_FEAT;        // N*64
  float* deg  = agg + (size_t)N * D_FEAT;        // N (becomes inv_deg)
  float* gsum = deg + N;                         // G*64
  float* gcnt = gsum + (size_t)G * D_FEAT;       // G

  const int T = 256;
  // in-degree -> inv_deg (layer-invariant)
  zero_kernel<<<(N + T - 1) / T, T, 0, stream>>>(deg, N);
  count_deg_kernel<<<(E + T - 1) / T, T, 0, stream>>>(dst, deg, E);
  inv_deg_kernel<<<(N + T - 1) / T, T, 0, stream>>>(deg, N);
  // h = x
  copy_kernel<<<(N * D_FEAT + T - 1) / T, T, 0, stream>>>(x, h, N * D_FEAT);

  const int etiles = E / 16, erem = E - etiles * 16;
  const int ntiles = N / 16, nrem = N - ntiles * 16;

  for (int l = 0; l < 3; ++l) {
    zero_kernel<<<(N * D_FEAT + T - 1) / T, T, 0, stream>>>(agg, N * D_FEAT);
    edge_kernel<<<(etiles + 7) / 8, T, 0, stream>>>(
        h, ea, src, dst,
        rw1 + (size_t)l * EDGE_DIM * HID, rb1 + (size_t)l * HID,
        rw2 + (size_t)l * HID * D_FEAT,   rb2 + (size_t)l * D_FEAT,
        agg, etiles);
    if (erem)
      edge_tail_kernel<<<1, 32, 0, stream>>>(
          h, ea, src, dst,
          rw1 + (size_t)l * EDGE_DIM * HID, rb1 + (size_t)l * HID,
          rw2 + (size_t)l * HID * D_FEAT,   rb2 + (size_t)l * D_FEAT,
          agg, etiles * 16, E);
    node_kernel<<<(ntiles + 7) / 8, T, 0, stream>>>(
        agg, deg, lw + (size_t)l * D_FEAT * D_FEAT, lb + (size_t)l * D_FEAT,
        h, ntiles);
    if (nrem)
      node_tail_kernel<<<1, 32, 0, stream>>>(
          agg, deg, lw + (size_t)l * D_FEAT * D_FEAT, lb + (size_t)l * D_FEAT,
          h, ntiles * 16, N);
  }

  zero_kernel<<<(G * D_FEAT + T - 1) / T, T, 0, stream>>>(gsum, G * D_FEAT);
  zero_kernel<<<1, T, 0, stream>>>(gcnt, G);
  pool_kernel<<<(N * D_FEAT + T - 1) / T, T, 0, stream>>>(h, batch, gsum, gcnt, N * D_FEAT);
  head_kernel<<<1, T, 0, stream>>>(gsum, gcnt, fw0, fb0, fw1, fb1, fw2, fb2, out, G);
}